// associative_loss_49830210568207
// MI455X (gfx1250) — compile-verified
//
#include <hip/hip_runtime.h>
#include <math.h>

typedef __attribute__((ext_vector_type(2))) float v2f;
typedef __attribute__((ext_vector_type(8))) float v8f;

#define EPSF 1e-8f

constexpr int Bn = 64;    // batches
constexpr int Tn = 2048;  // timesteps
constexpr int Dn = 256;   // feature dim
constexpr int Sn = 16;    // segments per pos/neg list

// ---------------------------------------------------------------------------
// Kernel 1: fused masked-matmul segment means via V_WMMA_F32_16X16X4_F32.
// One wave per (batch, 16-wide D tile). A-matrix (16x4 f32 mask) built in
// registers per the CDNA5 A layout with a single unsigned range-check per
// element (validity folded into the length); B-matrix (4x16 f32 feat tile)
// loaded non-temporally (feat is streamed exactly once; keep L2 for the
// means that kernel 2 re-reads). Pos and neg accumulators share every B
// tile, so feat is read once total -> bandwidth-bound (~134MB @ 23.3 TB/s).
// ---------------------------------------------------------------------------
__global__ __launch_bounds__(32)
void seg_mean_wmma(const float* __restrict__ feat,
                   const int*   __restrict__ idxp,
                   const int*   __restrict__ idxn,
                   float*       __restrict__ posM,
                   float*       __restrict__ negM)
{
    const int lane = threadIdx.x;        // 0..31 (wave32)
    const int b    = blockIdx.y;
    const int d0   = blockIdx.x * 16;
    const int m    = lane & 15;          // segment row this lane feeds into A
    const int half = lane >> 4;          // 0: K rows {0,1}, 1: K rows {2,3}

    const int* ip = idxp + b * (2 * Sn);
    const int* iq = idxn + b * (2 * Sn);

    const int sP = ip[2 * m], eP = ip[2 * m + 1];
    const int sN = iq[2 * m], eN = iq[2 * m + 1];

    // cumulative-validity (cumprod of starts >= 0) up to segment m
    bool vP = true, vN = true;
    for (int j = 0; j <= m; ++j) {
        vP = vP && (ip[2 * j] >= 0);
        vN = vN && (iq[2 * j] >= 0);
    }

    const int lP = (eP - sP > 1) ? (eP - sP) : 1;
    const int lN = (eN - sN > 1) ? (eN - sN) : 1;

    // mask(t) = (unsigned)(t - start) < uLen ; invalid segment -> uLen = 0
    const unsigned uLP = vP ? (unsigned)lP : 0u;
    const unsigned uLN = vN ? (unsigned)lN : 0u;

    __shared__ float sInvP[16], sInvN[16];
    if (lane < 16) {
        sInvP[lane] = 1.0f / (float)lP;
        sInvN[lane] = 1.0f / (float)lN;
    }
    __syncthreads();

    v8f accP = {};
    v8f accN = {};

    // column n = lane&15 of this D tile
    const float* fbase = feat + ((size_t)b * Tn) * Dn + d0 + (lane & 15);

    for (int kk = 0; kk < Tn; kk += 4) {
        const int t0 = kk + 2 * half;   // K rows handled by this lane
        const int t1 = t0 + 1;

        // A tiles (16x4 mask, f32): VGPR0 = K {0 | 2}, VGPR1 = K {1 | 3}
        v2f a_p, a_n, bv;
        a_p.x = ((unsigned)(t0 - sP) < uLP) ? 1.0f : 0.0f;
        a_p.y = ((unsigned)(t1 - sP) < uLP) ? 1.0f : 0.0f;
        a_n.x = ((unsigned)(t0 - sN) < uLN) ? 1.0f : 0.0f;
        a_n.y = ((unsigned)(t1 - sN) < uLN) ? 1.0f : 0.0f;

        // B tile (4x16 feat): same K-row split across half-waves.
        // Non-temporal: feat is single-use streaming data.
        const float* rp = fbase + (size_t)t0 * Dn;
        bv.x = __builtin_nontemporal_load(rp);
        bv.y = __builtin_nontemporal_load(rp + Dn);

        accP = __builtin_amdgcn_wmma_f32_16x16x4_f32(
            false, a_p, false, bv, (short)0, accP, false, false);
        accN = __builtin_amdgcn_wmma_f32_16x16x4_f32(
            false, a_n, false, bv, (short)0, accN, false, false);
    }

    // C/D layout: VGPR r -> M = r (lanes 0-15) / M = 8+r (lanes 16-31), N = lane&15
    float* op = posM + ((size_t)b * Sn) * Dn + d0 + (lane & 15);
    float* on = negM + ((size_t)b * Sn) * Dn + d0 + (lane & 15);
#pragma unroll
    for (int r = 0; r < 8; ++r) {
        const int M = r + 8 * half;
        op[(size_t)M * Dn] = accP[r] * sInvP[M];
        on[(size_t)M * Dn] = accN[r] * sInvN[M];
    }
}

// ---------------------------------------------------------------------------
// Block-wide deterministic tree reduction over 256 threads.
// ---------------------------------------------------------------------------
__device__ float blk_reduce(float v, float* sm)
{
    const int t = threadIdx.x;
    __syncthreads();
    sm[t] = v;
    __syncthreads();
    for (int off = 128; off > 0; off >>= 1) {
        if (t < off) sm[t] += sm[t + off];
        __syncthreads();
    }
    return sm[0];
}

// ---------------------------------------------------------------------------
// Kernel 2: per-batch center, cosines, loss1 + loss2. One block per batch,
// one thread per feature dim (D = 256).
// ---------------------------------------------------------------------------
__global__ __launch_bounds__(256)
void loss_kernel(const int*   __restrict__ idxp,
                 const int*   __restrict__ idxn,
                 const float* __restrict__ posM,
                 const float* __restrict__ negM,
                 float*       __restrict__ lossB)
{
    __shared__ float sm[256];
    const int b = blockIdx.x;
    const int d = threadIdx.x;

    const int* ip = idxp + b * (2 * Sn);
    const int* iq = idxn + b * (2 * Sn);

    bool vP[Sn], vN[Sn];
    int cP = 0, cN = 0;
    bool rp = true, rn = true;
#pragma unroll
    for (int s = 0; s < Sn; ++s) {
        rp = rp && (ip[2 * s] >= 0); vP[s] = rp; cP += rp ? 1 : 0;
        rn = rn && (iq[2 * s] >= 0); vN[s] = rn; cN += rn ? 1 : 0;
    }

    const float* pB = posM + ((size_t)b * Sn) * Dn + d;
    float pm[Sn];
    float cd = 0.0f;
#pragma unroll
    for (int s = 0; s < Sn; ++s) {
        pm[s] = pB[(size_t)s * Dn];
        cd += pm[s];
    }
    cd /= (float)cP;   // center[d]

    const float cn2 = blk_reduce(cd * cd, sm);
    const float cn  = fmaxf(sqrtf(cn2), EPSF);

    float loss1 = 0.0f;
    for (int s = 0; s < Sn; ++s) {
        const float dot = blk_reduce(pm[s] * cd, sm);
        const float n2  = blk_reduce(pm[s] * pm[s], sm);
        const float cv  = dot / (fmaxf(sqrtf(n2), EPSF) * cn);
        if (vP[s]) loss1 += 1.0f - cv;
    }
    loss1 /= (float)cP;

    const float* nB = negM + ((size_t)b * Sn) * Dn + d;
    float loss2 = 0.0f;
    for (int s = 0; s < Sn; ++s) {
        const float nm  = nB[(size_t)s * Dn];
        const float dot = blk_reduce(nm * cd, sm);
        const float n2  = blk_reduce(nm * nm, sm);
        const float cv  = dot / (fmaxf(sqrtf(n2), EPSF) * cn);
        if (vN[s]) loss2 += expf(-(1.0f - cv));
    }
    loss2 /= (float)cN;

    if (d == 0) lossB[b] = loss1 + loss2;
}

// ---------------------------------------------------------------------------
// Kernel 3: mean over batches -> scalar output.
// ---------------------------------------------------------------------------
__global__ __launch_bounds__(64)
void finalize_kernel(const float* __restrict__ lossB, float* __restrict__ out)
{
    __shared__ float sm[64];
    const int t = threadIdx.x;
    sm[t] = lossB[t];
    __syncthreads();
    for (int off = 32; off > 0; off >>= 1) {
        if (t < off) sm[t] += sm[t + off];
        __syncthreads();
    }
    if (t == 0) out[0] = sm[0] / (float)Bn;
}

extern "C" void kernel_launch(void* const* d_in, const int* in_sizes, int n_in,
                              void* d_out, int out_size, void* d_ws, size_t ws_size,
                              hipStream_t stream)
{
    const float* feat = (const float*)d_in[0];
    const int*   idxp = (const int*)d_in[1];
    const int*   idxn = (const int*)d_in[2];

    float* posM  = (float*)d_ws;                       // B*S*D floats (1 MiB)
    float* negM  = posM + (size_t)Bn * Sn * Dn;        // B*S*D floats (1 MiB)
    float* lossB = negM + (size_t)Bn * Sn * Dn;        // B floats

    dim3 grid(Dn / 16, Bn);
    seg_mean_wmma<<<grid, 32, 0, stream>>>(feat, idxp, idxn, posM, negM);
    loss_kernel<<<Bn, 256, 0, stream>>>(idxp, idxn, posM, negM, lossB);
    finalize_kernel<<<1, 64, 0, stream>>>(lossB, (float*)d_out);
}